// YoloLayer2_56693568307439
// MI455X (gfx1250) — compile-verified
//
#include <hip/hip_runtime.h>

// YOLO layer: (B, 3*85, 76, 76) f32 -> (B, 3, 76, 76, 85) f32
// Memory-bound transpose+pointwise (~189 MB total => ~8.1 us at 23.3 TB/s).
// Load phase: ONE Tensor Data Mover descriptor per workgroup DMAs the
// 85x76-float tile (rows of 304 B, stride 23104 B) straight into LDS.
// Store phase: fully contiguous, 16B-aligned float4 stores.

#define MASKS      3
#define ATTRIBS    85          // 5 + 80 classes
#define GH         76
#define GW         76
#define HW         (GH * GW)   // 5776
#define TILE_ELEMS (ATTRIBS * GW)       // 6460 floats = 25840 B of LDS
#define NCHUNK     (TILE_ELEMS / 4)     // 1615 x float4 output chunks

typedef unsigned int u32x4 __attribute__((ext_vector_type(4)));
typedef unsigned int u32x8 __attribute__((ext_vector_type(8)));

__global__ __launch_bounds__(256)
void yolo_head_kernel(const float* __restrict__ x, float* __restrict__ out) {
    __shared__ float tile[TILE_ELEMS];

    const int bid = blockIdx.x;        // 0 .. B*3*76-1
    const int row = bid % GH;          // j
    const int ba  = bid / GH;          // b*3 + a
    const int a   = ba % MASKS;

    // Input base for this (b, a, row): x[b][a*85 + 0][row][0]
    const float* gbase = x + (size_t)ba * ATTRIBS * HW + (size_t)row * GW;
    // Output base: out[b][a][row][0][0]  (76*85 contiguous floats, 16B aligned)
    float* obase = out + ((size_t)ba * HW + (size_t)row * GW) * ATTRIBS;

    // Low 32 bits of a flat pointer into LDS == LDS byte address.
    const unsigned lds0 = (unsigned)(size_t)(&tile[0]);
    const unsigned long long ga = (unsigned long long)(const void*)gbase;

    // ---- Tensor DMA descriptor (D#), per CDNA5 ISA ch.8 ----
    // Group 0: count=1 | lds_addr | global_addr[56:0] | type=2
    const u32x4 g0 = {
        1u,                                            // count = 1 valid descriptor
        lds0,                                          // LDS destination (bytes)
        (unsigned)ga,                                  // global_addr[31:0]
        (unsigned)((ga >> 32) & 0x01FFFFFFu) | 0x80000000u  // addr[56:32] | type=2<<30
    };
    // Group 1: data_size=4B, tensor 76x85, tile 76x85, dim0 stride 5776 elems
    const u32x8 g1 = {
        0x00020000u,          // workgroup_mask=0, data_size=2 (4B), no flags
        (unsigned)GW << 16,   // atomic_barrier_addr=0 | tensor_dim0[15:0]=76
        (unsigned)ATTRIBS << 16, // tensor_dim0[31:16]=0 | tensor_dim1[15:0]=85
        (unsigned)GW << 16,   // tensor_dim1[31:16]=0 | tile_dim0=76
        (unsigned)ATTRIBS,    // tile_dim1=85 | tile_dim2=0
        (unsigned)HW,         // tensor_dim0_stride[31:0] = 5776 (elements)
        0u,                   // stride0[47:32]=0 | tensor_dim1_stride[15:0]=0
        0u                    // tensor_dim1_stride[47:16]=0 (unused, tile_dim2=0)
    };

    // Exactly one wave issues the TDM op (scalar-uniform branch: tensor ops
    // ignore EXEC, so waves must branch around it, not merely mask lanes).
    if (__builtin_amdgcn_readfirstlane(threadIdx.x) == 0) {
        asm volatile("tensor_load_to_lds %0, %1"
                     :: "s"(g0), "s"(g1)
                     : "memory");
        __builtin_amdgcn_s_wait_tensorcnt(0);
    }
    __syncthreads();

    // ---- Compute phase: read LDS in output order, pointwise ops, float4 stores.
    const float inv76 = 1.0f / 76.0f;
    const float rowf  = (float)row;
    // Faithful to the reference's anchor indexing bug:
    // ANCHOR_W = anchors[0:3] = {10,13,16}, ANCHOR_H = anchors[1:4] = {13,16,30}
    const float AWs[MASKS] = {10.0f / 608.0f, 13.0f / 608.0f, 16.0f / 608.0f};
    const float AHs[MASKS] = {13.0f / 608.0f, 16.0f / 608.0f, 30.0f / 608.0f};
    const float aw = AWs[a];
    const float ah = AHs[a];

    for (int q = threadIdx.x; q < NCHUNK; q += 256) {
        float4 r4;
        float* rp = &r4.x;
#pragma unroll
        for (int c = 0; c < 4; ++c) {
            const int t = q * 4 + c;       // output-flat index = s*85 + k
            const int s = t / ATTRIBS;     // column i (0..75)
            const int k = t - s * ATTRIBS; // attrib (0..84)
            const float v = tile[k * GW + s];
            // numerically stable sigmoid: 1 / (1 + exp(-v))
            const float sg = __builtin_amdgcn_rcpf(1.0f + __expf(-v));
            float r;
            if (k >= 4) {
                r = sg;                                    // conf + class scores
            } else if (k == 0) {
                r = ((float)s + sg) * inv76;               // bx = (i + sig(tx))/W
            } else if (k == 1) {
                r = (rowf + sg) * inv76;                   // by = (j + sig(ty))/H
            } else {
                r = __expf(v) * (k == 2 ? aw : ah);        // bw / bh
            }
            rp[c] = r;
        }
        reinterpret_cast<float4*>(obase)[q] = r4;
    }
}

extern "C" void kernel_launch(void* const* d_in, const int* in_sizes, int n_in,
                              void* d_out, int out_size, void* d_ws, size_t ws_size,
                              hipStream_t stream) {
    const float* x = (const float*)d_in[0];
    float* out = (float*)d_out;

    const int B = in_sizes[0] / (MASKS * ATTRIBS * HW);   // 16 for the reference shapes
    const int nblocks = B * MASKS * GH;                    // one block per (b, a, row)

    yolo_head_kernel<<<dim3(nblocks), dim3(256), 0, stream>>>(x, out);
}